// GraphConvolution_31396210934417
// MI455X (gfx1250) — compile-verified
//
#include <hip/hip_runtime.h>
#include <hip/hip_bf16.h>
#include <math.h>

#define N_NODES 100000
#define F_DIM   512
#define D_DIM   128
#define NNZX    1600000
#define NNZE    1600000

typedef __attribute__((ext_vector_type(2))) float v2f;
typedef __attribute__((ext_vector_type(8))) float v8f;

// ---------------------------------------------------------------------------
// 0) Zero accumulators: pre_sup (ws), out (d_out), stats[0..259] (ws tail)
//    float4 stores: N*D divisible by 4; 102 MB of zeroing at full BW.
// ---------------------------------------------------------------------------
__global__ __launch_bounds__(256) void gcn_init_kernel(float4* __restrict__ pre_sup,
                                                       float4* __restrict__ out,
                                                       float*  __restrict__ stats) {
    const int stride = gridDim.x * blockDim.x;
    const int total4 = (N_NODES * D_DIM) / 4;
    const float4 z   = make_float4(0.f, 0.f, 0.f, 0.f);
    for (int i = blockIdx.x * blockDim.x + threadIdx.x; i < total4; i += stride) {
        pre_sup[i] = z;
        out[i]     = z;
    }
    if (blockIdx.x == 0) {
        for (int j = threadIdx.x; j < 260; j += blockDim.x) stats[j] = 0.0f;
    }
}

// ---------------------------------------------------------------------------
// 1) pre_sup = dropout(sparse X) @ W   (one wave per nnz, 4 cols per lane)
// ---------------------------------------------------------------------------
__global__ __launch_bounds__(256) void gcn_spmm_x_kernel(const float* __restrict__ xv,
                                                         const int*   __restrict__ xr,
                                                         const int*   __restrict__ xc,
                                                         const float* __restrict__ W,
                                                         const float* __restrict__ du,
                                                         float* __restrict__ pre_sup) {
    const int gid = blockIdx.x * 256 + threadIdx.x;
    const int i   = gid >> 5;
    if (i >= NNZX) return;
    // keep = floor(1-rate+u) == 1  <=>  u >= 0.5 ; rescale by 1/(1-rate) = 2
    const float u = du[i];
    if (u < 0.5f) return;                       // wave-uniform branch (i uniform)
    const int   lane = gid & 31;
    const float v    = xv[i] * 2.0f;
    const int   r    = xr[i];
    const int   c    = xc[i];
    const float4 w4  = *((const float4*)(W + (size_t)c * D_DIM) + lane);
    float* dst = pre_sup + (size_t)r * D_DIM + lane * 4;
    atomicAdd(dst + 0, v * w4.x);
    atomicAdd(dst + 1, v * w4.y);
    atomicAdd(dst + 2, v * w4.z);
    atomicAdd(dst + 3, v * w4.w);
}

// ---------------------------------------------------------------------------
// 2) out = sparse(support) @ pre_sup   (same COO pattern; gather + scatter)
// ---------------------------------------------------------------------------
__global__ __launch_bounds__(256) void gcn_spmm_e_kernel(const float* __restrict__ ev,
                                                         const int*   __restrict__ er,
                                                         const int*   __restrict__ ec,
                                                         const float* __restrict__ pre_sup,
                                                         float* __restrict__ out) {
    const int gid = blockIdx.x * 256 + threadIdx.x;
    const int i   = gid >> 5;
    if (i >= NNZE) return;
    const int   lane = gid & 31;
    const float v    = ev[i];
    const int   r    = er[i];
    const int   c    = ec[i];
    const float4 p4  = *((const float4*)(pre_sup + (size_t)c * D_DIM) + lane);
    float* dst = out + (size_t)r * D_DIM + lane * 4;
    atomicAdd(dst + 0, v * p4.x);
    atomicAdd(dst + 1, v * p4.y);
    atomicAdd(dst + 2, v * p4.z);
    atomicAdd(dst + 3, v * p4.w);
}

// ---------------------------------------------------------------------------
// 3) Column sum / sumsq via ones-GEMM on the f32 WMMA pipe, unrolled x4:
//    16 rows per iteration -> 8 clause'd loads, then 8 v_wmma_f32_16x16x4_f32.
//    A = 16x4 ones, B = 4x16 tile of out (resp. out^2). Since A is all-ones,
//    D[m,n] = sum_k B[k,n] for ANY bijective lane/vgpr->K placement, so the
//    only layout that must match hardware is the N mapping (n = lane%16,
//    same as the confirmed C/D layout). Every M row of D is identical ->
//    read element [0]. EXEC is all ones at every WMMA (no divergence).
// ---------------------------------------------------------------------------
__global__ __launch_bounds__(256) void gcn_stats_wmma_kernel(const float* __restrict__ out,
                                                             float* __restrict__ colsum,
                                                             float* __restrict__ colsq) {
    const int wave = threadIdx.x >> 5;        // 8 waves -> 8 column tiles (128 cols)
    const int lane = threadIdx.x & 31;
    const int col  = wave * 16 + (lane & 15);
    const int krow = (lane >> 4) * 2;         // half-wave covers rows {0,1} / {2,3}

    v2f a; a[0] = 1.0f; a[1] = 1.0f;          // ones A-matrix
    v8f csum = {};
    v8f csq  = {};

    const int nchunks = N_NODES / 16;         // 6250 iterations of 16 rows
    for (int chunk = blockIdx.x; chunk < nchunks; chunk += gridDim.x) {
        const float* base = out + (size_t)(chunk * 16 + krow) * D_DIM + col;
        // 8 loads -> one clause, one wait, then the WMMA chain
        const float t0 = base[0 * D_DIM], t1 = base[1 * D_DIM];
        const float t2 = base[4 * D_DIM], t3 = base[5 * D_DIM];
        const float t4 = base[8 * D_DIM], t5 = base[9 * D_DIM];
        const float t6 = base[12 * D_DIM], t7 = base[13 * D_DIM];

        v2f b0;  b0[0] = t0;       b0[1] = t1;
        v2f b1;  b1[0] = t2;       b1[1] = t3;
        v2f b2;  b2[0] = t4;       b2[1] = t5;
        v2f b3;  b3[0] = t6;       b3[1] = t7;
        v2f q0;  q0[0] = t0 * t0;  q0[1] = t1 * t1;
        v2f q1;  q1[0] = t2 * t2;  q1[1] = t3 * t3;
        v2f q2;  q2[0] = t4 * t4;  q2[1] = t5 * t5;
        v2f q3;  q3[0] = t6 * t6;  q3[1] = t7 * t7;

        csum = __builtin_amdgcn_wmma_f32_16x16x4_f32(false, a, false, b0, (short)0, csum, false, false);
        csq  = __builtin_amdgcn_wmma_f32_16x16x4_f32(false, a, false, q0, (short)0, csq,  false, false);
        csum = __builtin_amdgcn_wmma_f32_16x16x4_f32(false, a, false, b1, (short)0, csum, false, false);
        csq  = __builtin_amdgcn_wmma_f32_16x16x4_f32(false, a, false, q1, (short)0, csq,  false, false);
        csum = __builtin_amdgcn_wmma_f32_16x16x4_f32(false, a, false, b2, (short)0, csum, false, false);
        csq  = __builtin_amdgcn_wmma_f32_16x16x4_f32(false, a, false, q2, (short)0, csq,  false, false);
        csum = __builtin_amdgcn_wmma_f32_16x16x4_f32(false, a, false, b3, (short)0, csum, false, false);
        csq  = __builtin_amdgcn_wmma_f32_16x16x4_f32(false, a, false, q3, (short)0, csq,  false, false);
    }

    if (lane < 16) {                          // lanes 16-31 hold duplicate rows
        atomicAdd(&colsum[col], csum[0]);
        atomicAdd(&colsq[col],  csq[0]);
    }
}

// ---------------------------------------------------------------------------
// 4) BN (no scale/offset, biased var) + ReLU in place; accumulate sum(y^2)
// ---------------------------------------------------------------------------
__global__ __launch_bounds__(256) void gcn_bn_relu_kernel(float* __restrict__ out,
                                                          const float* __restrict__ colsum,
                                                          const float* __restrict__ colsq,
                                                          float* __restrict__ l2acc) {
    const int   stride = gridDim.x * blockDim.x;
    const int   total  = N_NODES * D_DIM;
    const float invN   = 1.0f / (float)N_NODES;
    float acc = 0.0f;
    for (int i = blockIdx.x * blockDim.x + threadIdx.x; i < total; i += stride) {
        const int   c    = i & (D_DIM - 1);
        const float mean = colsum[c] * invN;
        const float var  = colsq[c] * invN - mean * mean;
        float y = (out[i] - mean) * rsqrtf(var + 1.0e-3f);
        y = fmaxf(y, 0.0f);
        out[i] = y;
        acc = fmaf(y, y, acc);
    }
    for (int off = 16; off > 0; off >>= 1)    // wave32 reduction
        acc += __shfl_down(acc, off, 32);
    if ((threadIdx.x & 31) == 0) atomicAdd(l2acc, acc);
}

// ---------------------------------------------------------------------------
// 5) Global L2 normalization
// ---------------------------------------------------------------------------
__global__ __launch_bounds__(256) void gcn_l2_scale_kernel(float* __restrict__ out,
                                                           const float* __restrict__ l2acc) {
    const float s = rsqrtf(fmaxf(l2acc[0], 1.0e-12f));
    const int stride = gridDim.x * blockDim.x;
    const int total  = N_NODES * D_DIM;
    for (int i = blockIdx.x * blockDim.x + threadIdx.x; i < total; i += stride)
        out[i] *= s;
}

// ---------------------------------------------------------------------------
extern "C" void kernel_launch(void* const* d_in, const int* in_sizes, int n_in,
                              void* d_out, int out_size, void* d_ws, size_t ws_size,
                              hipStream_t stream) {
    const float* x_vals = (const float*)d_in[0];
    const int*   x_rows = (const int*)  d_in[1];
    const int*   x_cols = (const int*)  d_in[2];
    const float* e_vals = (const float*)d_in[3];
    const int*   e_rows = (const int*)  d_in[4];
    const int*   e_cols = (const int*)  d_in[5];
    const float* W      = (const float*)d_in[6];
    const float* du     = (const float*)d_in[7];

    float* out     = (float*)d_out;
    float* pre_sup = (float*)d_ws;                                 // N*D floats
    float* stats   = pre_sup + (size_t)N_NODES * D_DIM;            // 260 floats
    float* colsum  = stats;                                        // [128]
    float* colsq   = stats + 128;                                  // [128]
    float* l2acc   = stats + 256;                                  // [1]

    gcn_init_kernel<<<2048, 256, 0, stream>>>((float4*)pre_sup, (float4*)out, stats);
    gcn_spmm_x_kernel<<<(NNZX * 32) / 256, 256, 0, stream>>>(x_vals, x_rows, x_cols,
                                                             W, du, pre_sup);
    gcn_spmm_e_kernel<<<(NNZE * 32) / 256, 256, 0, stream>>>(e_vals, e_rows, e_cols,
                                                             pre_sup, out);
    gcn_stats_wmma_kernel<<<625, 256, 0, stream>>>(out, colsum, colsq);
    gcn_bn_relu_kernel<<<2048, 256, 0, stream>>>(out, colsum, colsq, l2acc);
    gcn_l2_scale_kernel<<<2048, 256, 0, stream>>>(out, l2acc);
}